// _PureMemoryEngineCore_40054865002733
// MI455X (gfx1250) — compile-verified
//
#include <hip/hip_runtime.h>
#include <hip/hip_bf16.h>

typedef float v2f __attribute__((ext_vector_type(2)));
typedef float v8f __attribute__((ext_vector_type(8)));
typedef unsigned int u32x4 __attribute__((ext_vector_type(4)));
typedef int i32x4 __attribute__((ext_vector_type(4)));
typedef int i32x8 __attribute__((ext_vector_type(8)));

#define MDIM 128
#define BDIM 8

#if defined(__has_builtin)
#if __has_builtin(__builtin_amdgcn_tensor_load_to_lds) && __has_builtin(__builtin_amdgcn_s_wait_tensorcnt)
#define HAVE_TDM 1
#endif
#endif

// ---- workspace float offsets ----
#define WS_G     0        // 128*128 complex interleaved (32768 floats), transient
#define WS_LTRE  32768    // L^T real  [j*128+i] = Re L[i][j]
#define WS_LTIM  49152    // L^T imag  (contiguous with LTRE: one 128KB TDM tile)
#define WS_SC    65536    // eta, directness, th_merge, th_prune
#define WS_RG    65540
#define WS_BR    65668
#define WS_RC    65796
#define WS_RS    65924
#define WS_T0R   66052
#define WS_T0I   66180
#define WS_WPRE  66308    // Wpred^T real, pair-packed [(k>>1)*256 + 2n + (k&1)]
#define WS_WPIM  82692    // Wpred^T imag, pair-packed
#define WS_OWT   99076    // out_w^T, pair-packed, K=256
// total = 131844 floats (~527KB)

// ---- dynamic LDS float offsets (scan kernel, no static LDS => base 0) ----
#define SM_LTRE  0
#define SM_LTIM  16384
#define SM_CRE   32768
#define SM_CIM   49152
#define SM_HB    65536    // 16x128 zero block (A-operand zero rows)
#define SM_H0    67584    // h_t double buffer (row 0 of A), 128
#define SM_H1    67712
#define SM_ARE   67840
#define SM_AIM   67968
#define SM_PRE   68096
#define SM_PIM   68224
#define SM_CURR  68352
#define SM_CURI  68480
#define SM_PRVR  68608
#define SM_PRVI  68736
#define SM_RED   68864    // 256
#define SM_REDJ  69120    // 256 (int)
#define SM_PB    69376    // prev staging: row0=re, row1=im (256)
#define SM_CAT   69632    // concat(re,im) staging for output chain (256)
#define SM_FLOATS 69888
#define SM_BYTES (SM_FLOATS * 4)

__device__ __forceinline__ float sigmoid_f(float x) { return 1.0f / (1.0f + __expf(-x)); }
__device__ __forceinline__ float softplus_f(float x) { return (x > 20.f) ? x : log1pf(__expf(x)); }

// ---------------------------------------------------------------------------
// Kernel A: Gram build (complex tanh), in-place parallel Gauss-Jordan inverse,
// store L^T planes + pair-packed Wpred^T / out_w^T + activation-baked params.
// ---------------------------------------------------------------------------
__global__ void pme_prep_kernel(const float* __restrict__ eps,
                                const float* __restrict__ eps_diag,
                                const float* __restrict__ tape_init,
                                const float* __restrict__ eta_raw,
                                const float* __restrict__ alpha,
                                const float* __restrict__ torque,
                                const float* __restrict__ w_r,
                                const float* __restrict__ bgate,
                                const float* __restrict__ tm_raw,
                                const float* __restrict__ tp_raw,
                                const float* __restrict__ wpred,
                                const float* __restrict__ outw,
                                float* __restrict__ ws)
{
    const int tid = threadIdx.x;
    float* G = ws + WS_G;
    const float scale = 0.35f / sqrtf((float)MDIM);

    for (int idx = tid; idx < MDIM * MDIM; idx += 256) {
        int i = idx >> 7, j = idx & 127;
        float gr, gi;
        if (i == j) {
            gr = 1.0f + 1e-3f + scale * tanhf(eps_diag[i]);
            gi = 0.0f;
        } else {
            float er = 0.5f * (eps[2 * (i * MDIM + j)]     + eps[2 * (j * MDIM + i)]);
            float ei = 0.5f * (eps[2 * (i * MDIM + j) + 1] - eps[2 * (j * MDIM + i) + 1]);
            float x = 2.f * er, y = 2.f * ei;
            float d = coshf(x) + __cosf(y);
            gr = scale * (sinhf(x) / d);
            gi = scale * (__sinf(y) / d);
        }
        G[2 * idx] = gr; G[2 * idx + 1] = gi;
    }
    __syncthreads();

    __shared__ float piv[2];
    __shared__ float fbuf[2 * MDIM];
    __shared__ float sred[256];

    for (int p = 0; p < MDIM; ++p) {
        if (tid == 0) {
            float ar = G[2 * (p * MDIM + p)], ai = G[2 * (p * MDIM + p) + 1];
            float den = ar * ar + ai * ai;
            piv[0] = ar / den; piv[1] = -ai / den;
        }
        __syncthreads();
        float dr = piv[0], di = piv[1];
        if (tid < MDIM && tid != p) {
            float ar = G[2 * (p * MDIM + tid)], ai = G[2 * (p * MDIM + tid) + 1];
            G[2 * (p * MDIM + tid)]     = ar * dr - ai * di;
            G[2 * (p * MDIM + tid) + 1] = ar * di + ai * dr;
        }
        if (tid < MDIM) {
            fbuf[2 * tid]     = G[2 * (tid * MDIM + p)];
            fbuf[2 * tid + 1] = G[2 * (tid * MDIM + p) + 1];
        }
        __syncthreads();
        {
            int r = tid >> 1, j0 = (tid & 1) * 64;
            if (r != p) {
                float fr = fbuf[2 * r], fi = fbuf[2 * r + 1];
                for (int jj = 0; jj < 64; ++jj) {
                    int j = j0 + jj;
                    if (j == p) continue;
                    float ar = G[2 * (p * MDIM + j)], ai = G[2 * (p * MDIM + j) + 1];
                    G[2 * (r * MDIM + j)]     -= fr * ar - fi * ai;
                    G[2 * (r * MDIM + j) + 1] -= fr * ai + fi * ar;
                }
                if (p >= j0 && p < j0 + 64) {
                    G[2 * (r * MDIM + p)]     = -(fr * dr - fi * di);
                    G[2 * (r * MDIM + p) + 1] = -(fr * di + fi * dr);
                }
            }
            if (tid == 0) { G[2 * (p * MDIM + p)] = dr; G[2 * (p * MDIM + p) + 1] = di; }
        }
        __syncthreads();
    }

    // L^T planes (LDS-destined, loaded once by TDM in the scan kernel)
    for (int idx = tid; idx < MDIM * MDIM; idx += 256) {
        int i = idx >> 7, j = idx & 127;
        ws[WS_LTRE + j * MDIM + i] = G[2 * idx];
        ws[WS_LTIM + j * MDIM + i] = G[2 * idx + 1];
    }
    // Wpred^T pair-packed planes: PK[(k>>1)*256 + 2n + (k&1)] = Wp[n][k]
    for (int idx = tid; idx < MDIM * MDIM; idx += 256) {
        int n = idx >> 7, k = idx & 127;
        int po = ((k >> 1) << 8) + 2 * n + (k & 1);
        ws[WS_WPRE + po] = wpred[2 * idx];
        ws[WS_WPIM + po] = wpred[2 * idx + 1];
    }
    // out_w^T pair-packed: k in [0,256)
    for (int idx = tid; idx < MDIM * 2 * MDIM; idx += 256) {
        int o = idx >> 8, k = idx & 255;
        ws[WS_OWT + ((k >> 1) << 8) + 2 * o + (k & 1)] = outw[idx];
    }

    if (tid < MDIM) {
        ws[WS_RG + tid] = sigmoid_f(w_r[tid]);
        ws[WS_BR + tid] = sigmoid_f(bgate[tid]);
        ws[WS_RC + tid] = __cosf(torque[tid]);
        ws[WS_RS + tid] = __sinf(torque[tid]);
    }
    if (tid == 0) {
        ws[WS_SC + 0] = softplus_f(eta_raw[0]);
        ws[WS_SC + 1] = sigmoid_f(alpha[0]);
        ws[WS_SC + 2] = softplus_f(tm_raw[0]);
        ws[WS_SC + 3] = softplus_f(tp_raw[0]);
    }
    float trr = 0.f, tii = 0.f;
    if (tid < MDIM) { trr = tape_init[2 * tid]; tii = tape_init[2 * tid + 1]; }
    sred[tid] = trr * trr + tii * tii;
    __syncthreads();
    for (int s = 128; s > 0; s >>= 1) { if (tid < s) sred[tid] += sred[tid + s]; __syncthreads(); }
    float nscl = 1.0f / fmaxf(sqrtf(sred[0]), 1e-8f);
    if (tid < MDIM) { ws[WS_T0R + tid] = trr * nscl; ws[WS_T0I + tid] = tii * nscl; }
}

// ---------------------------------------------------------------------------
// Kernel B: per-batch scan. corr + L^T LDS-resident (256KB of the 320KB WGP
// LDS). Three WMMA f32 16x16x4 chain groups per step:
//   reception a = h.L^T  (A row0 = h_t, async double-buffered via
//                         global_load_async_to_lds_b128)
//   pred = Wpred @ prev  (A row0 = prev_re, row1 = prev_im; B from packed ws)
//   out  = [re|im].outw^T (A row0 = concat, K=256; lanes store D directly)
// L^T loaded once via tensor_load_to_lds + s_wait_tensorcnt when available.
// ---------------------------------------------------------------------------
__global__ void pme_scan_kernel(const float* __restrict__ hid,
                                const float* __restrict__ ws,
                                float* __restrict__ out,
                                int S)
{
    extern __shared__ float sm[];
    const int tid = threadIdx.x;
    const int b   = blockIdx.x;

    float* Cre  = sm + SM_CRE;   float* Cim  = sm + SM_CIM;
    float* aRe  = sm + SM_ARE;   float* aIm  = sm + SM_AIM;
    float* pRe  = sm + SM_PRE;   float* pIm  = sm + SM_PIM;
    float* curR = sm + SM_CURR;  float* curI = sm + SM_CURI;
    float* prvR = sm + SM_PRVR;  float* prvI = sm + SM_PRVI;
    float* red  = sm + SM_RED;
    int*   redj = (int*)(sm + SM_REDJ);

    const float* wpre = ws + WS_WPRE;
    const float* wpim = ws + WS_WPIM;
    const float* owt  = ws + WS_OWT;
    const float* hrow = hid + (size_t)b * S * MDIM;

    const int lane = tid & 31;
    const int wv   = tid >> 5;
    const int n0   = wv * 16;
    const int koff = (lane < 16) ? 0 : 2;
    const int ln   = lane & 15;

    // ---- init: L^T via TDM (fallback: plain loads); zero corr + zero rows
#ifdef HAVE_TDM
    if (wv == 0) {
        unsigned long long ga = (unsigned long long)(ws + WS_LTRE);
        u32x4 g0;
        g0.x = 1u;                                   // count=1 valid descriptor
        g0.y = (unsigned)(SM_LTRE * 4);              // lds_addr (bytes)
        g0.z = (unsigned)(ga & 0xFFFFFFFFull);       // global_addr[31:0]
        g0.w = (unsigned)((ga >> 32) & 0x1FFFFFFull) | (2u << 30);  // addr[56:32] | type=2
        i32x8 g1;
        g1[0] = 0x20000;            // data_size=4B
        g1[1] = (int)0x80000000u;   // tensor_dim0 = 32768 (low16 << 16)
        g1[2] = 0x10000;            // tensor_dim0[31:16]=0 | tensor_dim1=1
        g1[3] = (int)0x80000000u;   // tile_dim0 = 32768
        g1[4] = 1;                  // tile_dim1 = 1
        g1[5] = 32768;              // tensor_dim0_stride
        g1[6] = 0; g1[7] = 0;
        i32x4 z4; z4[0] = 0; z4[1] = 0; z4[2] = 0; z4[3] = 0;
#if __clang_major__ >= 23
        i32x8 z8; z8[0]=0; z8[1]=0; z8[2]=0; z8[3]=0; z8[4]=0; z8[5]=0; z8[6]=0; z8[7]=0;
        __builtin_amdgcn_tensor_load_to_lds(g0, g1, z4, z4, z8, 0);
#else
        __builtin_amdgcn_tensor_load_to_lds(g0, g1, z4, z4, 0);
#endif
        __builtin_amdgcn_s_wait_tensorcnt(0);
    }
#else
    for (int idx = tid; idx < 2 * MDIM * MDIM; idx += 256) sm[SM_LTRE + idx] = ws[WS_LTRE + idx];
#endif
    for (int idx = tid; idx < MDIM * MDIM; idx += 256) { Cre[idx] = 0.f; Cim[idx] = 0.f; }
    for (int idx = tid; idx < 16 * MDIM; idx += 256) sm[SM_HB + idx] = 0.f;
    if (tid < MDIM) {
        sm[SM_H0 + tid] = hrow[tid];                 // synchronous h_0
        curR[tid] = ws[WS_T0R + tid]; curI[tid] = ws[WS_T0I + tid];
        prvR[tid] = curR[tid];        prvI[tid] = curI[tid];
    }
    const float eta    = ws[WS_SC + 0];
    const float direct = ws[WS_SC + 1];
    const float thm    = ws[WS_SC + 2];
    const float thp    = ws[WS_SC + 3];
    float rg = 0.f, br = 0.f, rc = 1.f, rs = 0.f;
    if (tid < MDIM) {
        rg = ws[WS_RG + tid]; br = ws[WS_BR + tid];
        rc = ws[WS_RC + tid]; rs = ws[WS_RS + tid];
    }
    __syncthreads();

    const unsigned long long hb64 = (unsigned long long)hrow;
    const int zrow = SM_HB + ln * MDIM;              // per-lane zero row base

    for (int t = 0; t < S; ++t) {
        // stage prev into A rows 0/1; kick async load of h_{t+1}
        if (tid < MDIM) { sm[SM_PB + tid] = prvR[tid]; sm[SM_PB + MDIM + tid] = prvI[tid]; }
        if (wv == 0 && t + 1 < S) {
            unsigned nb = ((t + 1) & 1) ? (unsigned)SM_H1 : (unsigned)SM_H0;
            unsigned ldsoff = nb * 4u + (unsigned)lane * 16u;
            unsigned voff   = (unsigned)(t + 1) * (MDIM * 4u) + (unsigned)lane * 16u;
            asm volatile("global_load_async_to_lds_b128 %0, %1, %2"
                         :: "v"(ldsoff), "v"(voff), "s"(hb64) : "memory");
        }
        if (t + 2 < S && tid < MDIM) __builtin_prefetch(&hrow[(size_t)(t + 2) * MDIM + tid], 0, 0);
        __syncthreads();

        // --- WMMA chains: reception (re/im) + pred (2 real chains)
        const int hbase = (ln == 0) ? (((t & 1) ? SM_H1 : SM_H0)) : zrow;
        const int pbase = (ln == 0) ? SM_PB : ((ln == 1) ? (SM_PB + MDIM) : zrow);
        v8f accR = {0.f,0.f,0.f,0.f,0.f,0.f,0.f,0.f};
        v8f accI = {0.f,0.f,0.f,0.f,0.f,0.f,0.f,0.f};
        v8f accP = {0.f,0.f,0.f,0.f,0.f,0.f,0.f,0.f};
        v8f accQ = {0.f,0.f,0.f,0.f,0.f,0.f,0.f,0.f};
        #pragma unroll 4
        for (int k0 = 0; k0 < MDIM; k0 += 4) {
            int k = k0 + koff;
            v2f afH, afP, bR, bI;
            afH.x = sm[hbase + k];               afH.y = sm[hbase + k + 1];
            afP.x = sm[pbase + k];               afP.y = sm[pbase + k + 1];
            bR.x  = sm[SM_LTRE + k * MDIM + n0 + ln];
            bR.y  = sm[SM_LTRE + (k + 1) * MDIM + n0 + ln];
            bI.x  = sm[SM_LTIM + k * MDIM + n0 + ln];
            bI.y  = sm[SM_LTIM + (k + 1) * MDIM + n0 + ln];
            int fo = ((k >> 1) << 8) + 2 * (n0 + ln);
            v2f bP = *(const v2f*)(wpre + fo);
            v2f bQ = *(const v2f*)(wpim + fo);
            accR = __builtin_amdgcn_wmma_f32_16x16x4_f32(false, afH, false, bR, (short)0, accR, false, false);
            accI = __builtin_amdgcn_wmma_f32_16x16x4_f32(false, afH, false, bI, (short)0, accI, false, false);
            accP = __builtin_amdgcn_wmma_f32_16x16x4_f32(false, afP, false, bP, (short)0, accP, false, false);
            accQ = __builtin_amdgcn_wmma_f32_16x16x4_f32(false, afP, false, bQ, (short)0, accQ, false, false);
        }
        if (lane < 16) {
            aRe[n0 + lane] = accR[0];
            aIm[n0 + lane] = accI[0];
            pRe[n0 + lane] = accP[0] - accQ[1];   // prev_re.WpRe^T - prev_im.WpIm^T
            pIm[n0 + lane] = accP[1] + accQ[0];   // prev_im.WpRe^T + prev_re.WpIm^T
        }
        __syncthreads();

        // --- state update + renorm
        float nr = 0.f, ni = 0.f;
        if (tid < MDIM) {
            int i = tid;
            float ar = aRe[i], ai = aIm[i], tr = curR[i], ti = curI[i];
            float rr = ar * tr - ai * ti, ri = ar * ti + ai * tr;
            float er = tr - pRe[i],       ei = ti - pIm[i];
            float cr = er * tr - ei * ti, ci = er * ti + ei * tr;
            float dr = direct * rr + 0.5f * cr;
            float di = direct * ri + 0.5f * ci;
            float qr = dr * rc - di * rs, qi = dr * rs + di * rc;
            nr = rg * tr + eta * br * qr;
            ni = rg * ti + eta * br * qi;
        }
        red[tid] = nr * nr + ni * ni;
        __syncthreads();
        for (int s = 128; s > 0; s >>= 1) { if (tid < s) red[tid] += red[tid + s]; __syncthreads(); }
        float scl = 1.0f / fmaxf(sqrtf(red[0]), 1e-8f);
        if (tid < MDIM) {
            prvR[tid] = curR[tid]; prvI[tid] = curI[tid];
            curR[tid] = nr * scl;  curI[tid] = ni * scl;
        }
        __syncthreads();

        // --- corr EMA
        {
            int i = tid >> 1, j0 = (tid & 1) * 64;
            float xr = curR[i], xi = curI[i];
            for (int jj = 0; jj < 64; ++jj) {
                int j = j0 + jj;
                float yr = curR[j], yi = curI[j];
                float opr = xr * yr + xi * yi;
                float opi = xi * yr - xr * yi;
                int idx = i * MDIM + j;
                Cre[idx] = 0.95f * Cre[idx] + 0.05f * opr;
                Cim[idx] = 0.95f * Cim[idx] + 0.05f * opi;
            }
        }
        __syncthreads();

        // --- consolidation every 8 steps
        if (((t + 1) & 7) == 0) {
            int i = tid >> 1, j0 = (tid & 1) * 64;
            float best = -1.f; int bj = 0;
            for (int jj = 0; jj < 64; ++jj) {
                int j = j0 + jj;
                if (j == i) continue;
                int idx = i * MDIM + j;
                float m2 = Cre[idx] * Cre[idx] + Cim[idx] * Cim[idx];
                if (m2 > best) { best = m2; bj = j; }
            }
            red[tid] = best; redj[tid] = bj;
            __syncthreads();
            float tr = 0.f, ti = 0.f, pr = 0.f, pi = 0.f, score = 0.f;
            if (tid < MDIM) {
                float m0 = red[2 * tid], m1 = red[2 * tid + 1];
                int pix = (m0 >= m1) ? redj[2 * tid] : redj[2 * tid + 1];
                score = sqrtf(fmaxf(m0, m1));
                tr = curR[tid]; ti = curI[tid];
                pr = curR[pix]; pi = curI[pix];
            }
            __syncthreads();
            if (tid < MDIM) {
                float mg = 0.15f * sigmoid_f((score - thm) * 8.f);
                float g  = 0.5f * mg;
                float mr = 0.5f * (tr + pr), mi = 0.5f * (ti + pi);
                tr = (1.f - g) * tr + g * mr;
                ti = (1.f - g) * ti + g * mi;
                float mag = sqrtf(tr * tr + ti * ti);
                float pg = sigmoid_f((thp - mag) * 8.f);
                float f2 = 1.f - 0.5f * pg;
                nr = tr * f2; ni = ti * f2;
            } else { nr = 0.f; ni = 0.f; }
            red[tid] = nr * nr + ni * ni;
            __syncthreads();
            for (int s = 128; s > 0; s >>= 1) { if (tid < s) red[tid] += red[tid + s]; __syncthreads(); }
            float s3 = 1.0f / fmaxf(sqrtf(red[0]), 1e-8f);
            if (tid < MDIM) { curR[tid] = nr * s3; curI[tid] = ni * s3; }
            __syncthreads();
        }

        // --- output projection via WMMA (K=256, A row0 = concat(re,im))
        if (tid < MDIM) { sm[SM_CAT + tid] = curR[tid]; sm[SM_CAT + MDIM + tid] = curI[tid]; }
        __syncthreads();
        {
            const int obase = (ln == 0) ? SM_CAT : (SM_HB + MDIM);
            v8f accO = {0.f,0.f,0.f,0.f,0.f,0.f,0.f,0.f};
            #pragma unroll 4
            for (int k0 = 0; k0 < 2 * MDIM; k0 += 4) {
                int k = k0 + koff;
                v2f afO;
                afO.x = sm[obase + k]; afO.y = sm[obase + k + 1];
                v2f bO = *(const v2f*)(owt + ((k >> 1) << 8) + 2 * (n0 + ln));
                accO = __builtin_amdgcn_wmma_f32_16x16x4_f32(false, afO, false, bO, (short)0, accO, false, false);
            }
            if (lane < 16) out[((size_t)b * S + t) * MDIM + n0 + lane] = accO[0];
        }

        if (wv == 0) asm volatile("s_wait_asynccnt 0" ::: "memory");
        __syncthreads();
    }
}

extern "C" void kernel_launch(void* const* d_in, const int* in_sizes, int n_in,
                              void* d_out, int out_size, void* d_ws, size_t ws_size,
                              hipStream_t stream) {
    (void)n_in; (void)out_size; (void)ws_size;
    const float* hid      = (const float*)d_in[0];
    const float* outw     = (const float*)d_in[1];
    const float* tape0    = (const float*)d_in[2];
    const float* eta_raw  = (const float*)d_in[3];
    const float* alpha    = (const float*)d_in[4];
    const float* torque   = (const float*)d_in[5];
    const float* eps      = (const float*)d_in[6];
    const float* eps_diag = (const float*)d_in[7];
    const float* wpred    = (const float*)d_in[8];
    const float* w_r      = (const float*)d_in[9];
    const float* bgate    = (const float*)d_in[10];
    const float* tm_raw   = (const float*)d_in[11];
    const float* tp_raw   = (const float*)d_in[12];
    float* ws  = (float*)d_ws;
    float* out = (float*)d_out;

    int S = in_sizes[0] / (BDIM * MDIM);   // 1024

    (void)hipFuncSetAttribute((const void*)pme_scan_kernel,
                              hipFuncAttributeMaxDynamicSharedMemorySize, SM_BYTES);

    pme_prep_kernel<<<1, 256, 0, stream>>>(eps, eps_diag, tape0, eta_raw, alpha, torque,
                                           w_r, bgate, tm_raw, tp_raw, wpred, outw, ws);
    pme_scan_kernel<<<BDIM, 256, SM_BYTES, stream>>>(hid, ws, out, S);
}